// OneHotEncoding_83408264888701
// MI455X (gfx1250) — compile-verified
//
#include <hip/hip_runtime.h>
#include <hip/hip_bf16.h>

// out[t, d] = relu(W[d, poses[t]] + b[d])
//   poses: [16384] int, W: [256, 100000] f32 row-major, b: [256] f32
//   out:   [16384, 256] f32
//
// Memory-bound gather. W (102.4 MB) fits in MI455X's 192 MB L2; output is
// streamed with non-temporal stores so it never evicts W. 16 tokens per
// 256-thread block -> 16 outstanding gather loads per lane (deep MLP),
// one s_load_b512 for the uniform indices, and pure 32-bit GVS addressing
// (saddr64 + vaddr32 * scale_offset) since all byte offsets fit in u32.

#define N_CLASSES 100000u
#define DIM_EMB   256u
#define N_TOKENS  16384u
#define TPB       16       // tokens per block
#define BLOCK     256      // 8 wave32 waves

__global__ __launch_bounds__(BLOCK) void
OneHotEmbed_gather_kernel(const int* __restrict__ poses,
                          const float* __restrict__ W,
                          const float* __restrict__ b,
                          float* __restrict__ out) {
    const unsigned d  = threadIdx.x;            // output dim owned by this lane
    const unsigned t0 = blockIdx.x * TPB;       // first token of this block

    // Per-thread bias, reused across all TPB tokens.
    const float bias = b[d];

    // Element offset of W[d, 0]; d*100000 <= 25.6M elements -> fits u32,
    // byte offset <= 102.8 MB -> GVS scale_offset addressing, no 64-bit VALU.
    const unsigned wbase = d * N_CLASSES;

    // Block-uniform index loads -> one s_load_b512 (KMcnt), keeping the
    // vector-memory pipe free for the W gather. Unsigned: kills the
    // per-token sign-extend + 64-bit add chains seen in the previous build.
    unsigned pos[TPB];
#pragma unroll
    for (int k = 0; k < TPB; ++k) {
        pos[k] = (unsigned)poses[t0 + k];
    }

    // Issue all TPB gathers back-to-back before consuming any: 16
    // outstanding global_load_b32 per lane (LOADcnt-tracked, cap 63) to
    // hide L2 gather latency. W lines stay RT-cached (hot in 192 MB L2).
    float v[TPB];
#pragma unroll
    for (int k = 0; k < TPB; ++k) {
        v[k] = W[wbase + pos[k]];
    }

    // bias + relu, then coalesced non-temporal stores (TH=NT): out is
    // write-once / never re-read — don't let it evict the L2-resident W.
    // Max element offset 4.19M -> u32 GVS addressing for stores too.
    const unsigned obase = t0 * DIM_EMB + d;
#pragma unroll
    for (int k = 0; k < TPB; ++k) {
        const float r = fmaxf(v[k] + bias, 0.0f);
        __builtin_nontemporal_store(r, &out[obase + (unsigned)k * DIM_EMB]);
    }
}

extern "C" void kernel_launch(void* const* d_in, const int* in_sizes, int n_in,
                              void* d_out, int out_size, void* d_ws, size_t ws_size,
                              hipStream_t stream) {
    (void)in_sizes; (void)n_in; (void)out_size; (void)d_ws; (void)ws_size;

    const int*   poses = (const int*)  d_in[0];   // [N_TOKENS]
    const float* W     = (const float*)d_in[1];   // [DIM_EMB, N_CLASSES]
    const float* b     = (const float*)d_in[2];   // [DIM_EMB]
    float*       out   = (float*)      d_out;     // [N_TOKENS, DIM_EMB]

    const int grid = N_TOKENS / TPB;              // 1024 blocks of 8 waves
    OneHotEmbed_gather_kernel<<<grid, BLOCK, 0, stream>>>(poses, W, b, out);
}